// wvfn_2688649527963
// MI455X (gfx1250) — compile-verified
//
#include <hip/hip_runtime.h>
#include <stdint.h>

// ---- problem constants (match reference) ----
#define NCOORD 6
#define FLOATS_PER_WALKER (NCOORD * 3)   // 18 floats = 72 B
#define WPB 256                          // walkers per block = threads per block
#define N128_FULL (WPB * FLOATS_PER_WALKER / 4)   // 1152 16B-chunks per full block

typedef int v2i __attribute__((ext_vector_type(2)));
typedef int v4i __attribute__((ext_vector_type(4)));
typedef __attribute__((address_space(1))) void GV;
typedef __attribute__((address_space(3))) void LV;
typedef __attribute__((address_space(1))) v2i Gv2i;
typedef __attribute__((address_space(3))) v2i Lv2i;
typedef __attribute__((address_space(1))) v4i Gv4i;
typedef __attribute__((address_space(3))) v4i Lv4i;

__device__ __forceinline__ void async_copy_b128(const float* gsrc, float* lds_dst) {
#if __has_builtin(__builtin_amdgcn_global_load_async_to_lds_b128)
    __builtin_amdgcn_global_load_async_to_lds_b128(
        (Gv4i*)(GV*)gsrc, (Lv4i*)(LV*)lds_dst, /*offset=*/0, /*cpol=*/0);
#else
    unsigned lds_off = (unsigned)(uintptr_t)(LV*)lds_dst;
    unsigned long long ga = (unsigned long long)(uintptr_t)gsrc;
    asm volatile("global_load_async_to_lds_b128 %0, %1, off"
                 :: "v"(lds_off), "v"(ga) : "memory");
#endif
}

__device__ __forceinline__ void async_copy_b64(const float* gsrc, float* lds_dst) {
#if __has_builtin(__builtin_amdgcn_global_load_async_to_lds_b64)
    __builtin_amdgcn_global_load_async_to_lds_b64(
        (Gv2i*)(GV*)gsrc, (Lv2i*)(LV*)lds_dst, /*offset=*/0, /*cpol=*/0);
#else
    unsigned lds_off = (unsigned)(uintptr_t)(LV*)lds_dst;
    unsigned long long ga = (unsigned long long)(uintptr_t)gsrc;
    asm volatile("global_load_async_to_lds_b64 %0, %1, off"
                 :: "v"(lds_off), "v"(ga) : "memory");
#endif
}

__device__ __forceinline__ void wait_async_zero() {
#if __has_builtin(__builtin_amdgcn_s_wait_asynccnt)
    __builtin_amdgcn_s_wait_asynccnt(0);
#else
    asm volatile("s_wait_asynccnt 0" ::: "memory");
#endif
}

// rsqrt via v_rsq_f32 + one Newton-Raphson step: ~1-2 ulp, ~6 ops vs ~14 for sqrt+div
__device__ __forceinline__ float fast_rsqrt(float r2) {
#if __has_builtin(__builtin_amdgcn_rsqf)
    float iv = __builtin_amdgcn_rsqf(r2);
#else
    float iv = rsqrtf(r2);
#endif
    return iv * (1.5f - 0.5f * r2 * iv * iv);
}

__global__ __launch_bounds__(WPB) void wvfn_kernel(
    const float* __restrict__ Rs,   // [n, 6, 3] f32
    const float* __restrict__ A,    // [6] f32 (only A[0] used)
    const float* __restrict__ C,    // [6] complex64 -> 12 floats (only C[0] used)
    float* __restrict__ out,        // [2n] interleaved complex out1, then [n] out2
    int n)
{
    __shared__ alignas(16) float smem[WPB * FLOATS_PER_WALKER];   // 18432 B

    const int t = threadIdx.x;
    const int blockBase = blockIdx.x * WPB;
    int nvalid = n - blockBase;
    if (nvalid > WPB) nvalid = WPB;

    // ---- coalesced async global -> LDS staging (ASYNCcnt path) ----
    const float* gbase = Rs + (size_t)blockBase * FLOATS_PER_WALKER;
    if (nvalid == WPB) {
        // fast path: 1152 16B chunks, k=0..3 unconditional, k=4 half-populated
#pragma unroll
        for (int k = 0; k < 4; ++k) {
            int c = t + k * WPB;
            async_copy_b128(gbase + 4 * c, &smem[4 * c]);
        }
        int c4 = t + 4 * WPB;
        if (c4 < N128_FULL) {
            async_copy_b128(gbase + 4 * c4, &smem[4 * c4]);
        }
    } else {
        const int totalBytes = nvalid * (FLOATS_PER_WALKER * 4);
        const int n128 = totalBytes >> 4;
#pragma unroll
        for (int k = 0; k < 5; ++k) {
            int c = t + k * WPB;
            if (c < n128) {
                async_copy_b128(gbase + 4 * c, &smem[4 * c]);
            }
        }
        if ((totalBytes & 15) && t == 0) {            // 8B remainder (odd nvalid)
            async_copy_b64(gbase + 4 * n128, &smem[4 * n128]);
        }
    }
    wait_async_zero();
    __syncthreads();

    const int w = blockBase + t;
    if (w >= n) return;

    // ---- per-lane physics (conflict-free LDS reads: 18*l mod 64 bijective) ----
    const float* p = &smem[t * FLOATS_PER_WALKER];
    float x[NCOORD], y[NCOORD], z[NCOORD];
#pragma unroll
    for (int i = 0; i < NCOORD; ++i) {
        x[i] = p[3 * i + 0];
        y[i] = p[3 * i + 1];
        z[i] = p[3 * i + 2];
    }

    float S = 0.0f, isum = 0.0f;
    float gx[NCOORD], gy[NCOORD], gz[NCOORD];
#pragma unroll
    for (int i = 0; i < NCOORD; ++i) { gx[i] = 0.0f; gy[i] = 0.0f; gz[i] = 0.0f; }

#pragma unroll
    for (int i = 0; i < NCOORD; ++i) {
#pragma unroll
        for (int j = i + 1; j < NCOORD; ++j) {
            float dx = x[i] - x[j];
            float dy = y[i] - y[j];
            float dz = z[i] - z[j];
            float r2 = dx * dx + dy * dy + dz * dz;
            float iv = fast_rsqrt(r2);     // 1/r
            float r  = r2 * iv;            // r
            S    += r;
            isum += iv;
            float ex = dx * iv, ey = dy * iv, ez = dz * iv;
            gx[i] += ex; gy[i] += ey; gz[i] += ez;
            gx[j] -= ex; gy[j] -= ey; gz[j] -= ez;
        }
    }

    float g2 = 0.0f;
#pragma unroll
    for (int i = 0; i < NCOORD; ++i)
        g2 += gx[i] * gx[i] + gy[i] * gy[i] + gz[i] * gz[i];

    const float VB = 1.0f;                 // ALPHA * CF = 0.75 * 4/3
    float a0 = A[0];
    float ia0 = 1.0f / a0;                 // single precise divide per walker
    float inv_ord = 2.0f * isum;           // ordered-pair sum of 1/r
    float lap  = g2 * ia0 * ia0 - 2.0f * ia0 * inv_ord;
    float hfac = -0.5f * lap - 0.5f * VB * inv_ord;   // K + V factor (real)

    float e   = expf(-S * ia0);
    float c0r = C[0], c0i = C[1];
    float pr = c0r * e, pi = c0i * e;

    // conj(psi) * (psi * hfac) / VB^2   (VB == 1)
    float hre = pr * hfac, him = pi * hfac;
    float o1re = pr * hre + pi * him;
    float o1im = pr * him - pi * hre;
    float o2   = pr * pr + pi * pi;

    out[2 * (size_t)w + 0] = o1re;
    out[2 * (size_t)w + 1] = o1im;
    out[2 * (size_t)n + w] = o2;
}

extern "C" void kernel_launch(void* const* d_in, const int* in_sizes, int n_in,
                              void* d_out, int out_size, void* d_ws, size_t ws_size,
                              hipStream_t stream) {
    const float* Rs = (const float*)d_in[0];   // f32 [W,6,3]
    const float* A  = (const float*)d_in[1];   // f32 [6]
    const float* C  = (const float*)d_in[2];   // complex64 [6] -> float pairs
    float* out = (float*)d_out;

    int n = in_sizes[0] / FLOATS_PER_WALKER;
    int blocks = (n + WPB - 1) / WPB;
    hipLaunchKernelGGL(wvfn_kernel, dim3(blocks), dim3(WPB), 0, stream,
                       Rs, A, C, out, n);
}